// PerMolSyner_56470230008503
// MI455X (gfx1250) — compile-verified
//
#include <hip/hip_runtime.h>
#include <hip/hip_bf16.h>
#include <math.h>

// ---------------------------------------------------------------------------
// PerMolSyner forward for gfx1250 (MI455X, wave32, WMMA).
// All GEMM-shaped compute uses V_WMMA_F32_16X16X4_F32 (fp32 matrix pipe).
// Attention is fused flash-style (QK^T + online softmax + P*V, both WMMA).
// Conv1d(k=9) is lowered to im2col + WMMA GEMM.
//
// Input flattening assumption (recursive insertion order of setup_inputs()):
//   0:x1 1:x1_sm 2:x2 3:x2_sm 4:cell
//   5..38: params scalars-in-order (fc_W,fc_b,g1_W,g1_b,g2_W,g2_b,g3_W,g3_b,
//          fcg1_W,fcg1_b,fcg2_W,fcg2_b,r1_W,r1_b,r2_W,r2_b,r3_W,r3_b,
//          d1_W,d1_b,d2_W,d2_b,d3_W,d3_b,map1_W,map1_b,map2_W,map2_b,
//          fc1_W,fc1_b,fc2_W,fc2_b,out_W,out_b)
//   39..48: enc_prot (fc_W,fc_b, convW0,convb0, convW1,convb1, convW2,convb2, ln_g,ln_b)
//   49..58: enc_smi  (same layout)
//   59..126: dec_smi  (ft_W,ft_b, then 3 layers x [ln_g,ln_b, sa(wq,bq,wk,bk,wv,bv,fo,bo),
//                      ea(8), pf_w1,pf_b1,pf_w2,pf_b2])
//   127..194: dec_prot (same layout)
//   195:edge_index1 196:edge_index2 197:batch1 198:batch2 199:new_batch1 200:new_batch2
// ---------------------------------------------------------------------------

typedef __attribute__((ext_vector_type(2))) float wf2;
typedef __attribute__((ext_vector_type(8))) float wf8;

#define DEVFN __device__ __forceinline__

DEVFN unsigned ord_enc(float f) {
  unsigned u = __float_as_uint(f);
  return (u & 0x80000000u) ? ~u : (u | 0x80000000u);
}
DEVFN float ord_dec(unsigned u) {
  return __uint_as_float((u & 0x80000000u) ? (u & 0x7fffffffu) : ~u);
}

// ---------------------------------------------------------------------------
// Generic fp32 WMMA GEMM:  Cout = (Cin? Cin:0) + A[MxK] @ B[KxN] + bias, act
// One wave computes one 16x16 output tile; 8 waves / 256-thread block.
// ---------------------------------------------------------------------------
#define GEMM_WAVES 8

__global__ __launch_bounds__(256) void gemm_wmma_kernel(
    const float* __restrict__ A, const float* __restrict__ B,
    const float* __restrict__ bias, const float* __restrict__ Cin,
    float* __restrict__ Cout, int M, int N, int K, int act) {
  int wave = threadIdx.x >> 5;
  int tn_count = (N + 15) >> 4;
  int tm_count = (M + 15) >> 4;
  int tile = blockIdx.x * GEMM_WAVES + wave;
  if (tile >= tm_count * tn_count) return;
  int tm = tile / tn_count;
  int tn = tile - tm * tn_count;
  int lane = threadIdx.x & 31;
  int half = lane >> 4;
  int l16 = lane & 15;
  int row0 = tm * 16, col0 = tn * 16;
  int ar = row0 + l16;
  int arc = min(ar, M - 1);          // clamp (garbage rows never stored)
  int bc = col0 + l16;
  int bcc = min(bc, N - 1);          // clamp (garbage cols never stored)
  const float* Ap = A + (size_t)arc * K;

  wf8 acc;
  if (Cin) {
#pragma unroll
    for (int v = 0; v < 8; ++v) {
      int r = row0 + v + half * 8;
      acc[v] = (r < M && bc < N) ? Cin[(size_t)r * N + bc] : 0.f;
    }
  } else {
#pragma unroll
    for (int v = 0; v < 8; ++v) acc[v] = 0.f;
  }

  int Kfull = K & ~3;
  int k0 = 0;
  for (; k0 < Kfull; k0 += 4) {
    int ak = k0 + half * 2;
    wf2 a, b;
    a.x = Ap[ak];
    a.y = Ap[ak + 1];
    b.x = B[(size_t)ak * N + bcc];
    b.y = B[(size_t)(ak + 1) * N + bcc];
    acc = __builtin_amdgcn_wmma_f32_16x16x4_f32(false, a, false, b, (short)0,
                                                acc, false, false);
  }
  if (k0 < K) {  // K tail (1..3), zero-pad
    int ak = k0 + half * 2;
    wf2 a, b;
    a.x = (ak < K) ? Ap[ak] : 0.f;
    a.y = (ak + 1 < K) ? Ap[ak + 1] : 0.f;
    b.x = (ak < K) ? B[(size_t)ak * N + bcc] : 0.f;
    b.y = (ak + 1 < K) ? B[(size_t)(ak + 1) * N + bcc] : 0.f;
    acc = __builtin_amdgcn_wmma_f32_16x16x4_f32(false, a, false, b, (short)0,
                                                acc, false, false);
  }

  if (bc < N) {
    float bv = bias ? bias[bc] : 0.f;
#pragma unroll
    for (int v = 0; v < 8; ++v) {
      int r = row0 + v + half * 8;
      if (r < M) {
        float xv = acc[v] + bv;
        if (act == 1) xv = fmaxf(xv, 0.f);
        Cout[(size_t)r * N + bc] = xv;
      }
    }
  }
}

// ---------------------------------------------------------------------------
// Fused flash attention, H=8, dh=16, one wave per (head, 16-query tile).
// Q,K,V,O are [L,128] with head h occupying columns h*16..h*16+15.
// ---------------------------------------------------------------------------
__global__ __launch_bounds__(32) void attn_flash_kernel(
    const float* __restrict__ Q, const float* __restrict__ K,
    const float* __restrict__ V, float* __restrict__ O, int Lq, int Lk) {
  __shared__ float ldsP[16 * 17];
  const int h = blockIdx.y;
  const int q0 = blockIdx.x * 16;
  const int lane = threadIdx.x & 31;
  const int half = lane >> 4;
  const int l16 = lane & 15;
  const int cbase = h * 16;

  wf2 qa[4];
#pragma unroll
  for (int s = 0; s < 4; ++s) {
    int k = s * 4 + half * 2;
    const float* qrow = Q + (size_t)(q0 + l16) * 128 + cbase;
    qa[s].x = qrow[k];
    qa[s].y = qrow[k + 1];
  }

  wf8 acc;
  float mrow[8], lrow[8];
#pragma unroll
  for (int v = 0; v < 8; ++v) {
    acc[v] = 0.f;
    mrow[v] = -3.402823466e38f;
    lrow[v] = 0.f;
  }

  for (int kt = 0; kt < Lk; kt += 16) {
    // S = Q @ K^T (dh=16 -> 4 wmma steps)
    wf8 sc;
#pragma unroll
    for (int v = 0; v < 8; ++v) sc[v] = 0.f;
#pragma unroll
    for (int s = 0; s < 4; ++s) {
      int k = s * 4 + half * 2;
      const float* krow = K + (size_t)(kt + l16) * 128 + cbase;
      wf2 b;
      b.x = krow[k];
      b.y = krow[k + 1];
      sc = __builtin_amdgcn_wmma_f32_16x16x4_f32(false, qa[s], false, b,
                                                 (short)0, sc, false, false);
    }
    // online softmax; rows of this half live in lanes sharing (lane>>4)
#pragma unroll
    for (int v = 0; v < 8; ++v) {
      float sl = sc[v] * 0.25f;  // 1/sqrt(dh=16)
      float tmax = sl;
#pragma unroll
      for (int m = 1; m <= 8; m <<= 1)
        tmax = fmaxf(tmax, __shfl_xor(tmax, m, 32));
      float mn = fmaxf(mrow[v], tmax);
      float p = __expf(sl - mn);
      float rs = p;
#pragma unroll
      for (int m = 1; m <= 8; m <<= 1) rs += __shfl_xor(rs, m, 32);
      float corr = __expf(mrow[v] - mn);
      lrow[v] = lrow[v] * corr + rs;
      acc[v] = acc[v] * corr;
      mrow[v] = mn;
      ldsP[(v + half * 8) * 17 + l16] = p;
    }
    __syncthreads();
    // acc += P @ V_tile (K=16 -> 4 wmma steps); P relayout via LDS
#pragma unroll
    for (int s = 0; s < 4; ++s) {
      int k = s * 4 + half * 2;
      wf2 a, b;
      a.x = ldsP[l16 * 17 + k];
      a.y = ldsP[l16 * 17 + k + 1];
      const float* vrow = V + (size_t)(kt + k) * 128 + cbase + l16;
      b.x = vrow[0];
      b.y = vrow[128];
      acc = __builtin_amdgcn_wmma_f32_16x16x4_f32(false, a, false, b, (short)0,
                                                  acc, false, false);
    }
    __syncthreads();
  }
#pragma unroll
  for (int v = 0; v < 8; ++v) {
    int row = q0 + v + half * 8;
    O[(size_t)row * 128 + cbase + l16] = acc[v] / lrow[v];
  }
}

// ---------------------------------------------------------------------------
// LayerNorm over D=128, optional residual; one wave per row.
// ---------------------------------------------------------------------------
__global__ __launch_bounds__(32) void layernorm_kernel(
    const float* __restrict__ x, const float* __restrict__ res,
    const float* __restrict__ g, const float* __restrict__ b,
    float* __restrict__ out) {
  int row = blockIdx.x;
  int lane = threadIdx.x;
  const float* xr = x + (size_t)row * 128;
  float vals[4];
  float sum = 0.f;
#pragma unroll
  for (int i = 0; i < 4; ++i) {
    float v = xr[lane + i * 32];
    if (res) v += res[(size_t)row * 128 + lane + i * 32];
    vals[i] = v;
    sum += v;
  }
#pragma unroll
  for (int m = 1; m <= 16; m <<= 1) sum += __shfl_xor(sum, m, 32);
  float mean = sum * (1.f / 128.f);
  float var = 0.f;
#pragma unroll
  for (int i = 0; i < 4; ++i) {
    float d = vals[i] - mean;
    var += d * d;
  }
#pragma unroll
  for (int m = 1; m <= 16; m <<= 1) var += __shfl_xor(var, m, 32);
  float inv = rsqrtf(var * (1.f / 128.f) + 1e-5f);
#pragma unroll
  for (int i = 0; i < 4; ++i) {
    int c = lane + i * 32;
    out[(size_t)row * 128 + c] = (vals[i] - mean) * inv * g[c] + b[c];
  }
}

// ---------------------------------------------------------------------------
// Encoder helpers: im2col for conv1d(k=9,pad=4) over [L,128]; weight repack.
// ---------------------------------------------------------------------------
__global__ void im2col_kernel(const float* __restrict__ h,
                              float* __restrict__ col, int L) {
  int idx = blockIdx.x * blockDim.x + threadIdx.x;
  int total = L * 1152;  // 9*128
  if (idx >= total) return;
  int l = idx / 1152;
  int rem = idx - l * 1152;
  int t = rem >> 7;        // 0..8
  int ic = rem & 127;
  int sl = l + t - 4;
  col[idx] = (sl >= 0 && sl < L) ? h[(size_t)sl * 128 + ic] : 0.f;
}

__global__ void repack_w_kernel(const float* __restrict__ W,
                                float* __restrict__ Wr) {
  int idx = blockIdx.x * blockDim.x + threadIdx.x;  // O*C*Kw = 256*128*9
  if (idx >= 256 * 128 * 9) return;
  int oc = idx / (128 * 9);
  int rem = idx - oc * (128 * 9);
  int ic = rem / 9;
  int t = rem - ic * 9;
  Wr[(size_t)(t * 128 + ic) * 256 + oc] = W[idx];
}

__global__ void glu_res_kernel(const float* __restrict__ conv,
                               float* __restrict__ ci, int L) {
  int idx = blockIdx.x * blockDim.x + threadIdx.x;
  if (idx >= L * 128) return;
  int l = idx >> 7;
  int c = idx & 127;
  float a = conv[(size_t)l * 256 + c];
  float g = conv[(size_t)l * 256 + 128 + c];
  float sg = 1.f / (1.f + __expf(-g));
  ci[idx] = (a * sg + ci[idx]) * 0.70710678118f;
}

// ---------------------------------------------------------------------------
// Elementwise / GCN / pooling helpers
// ---------------------------------------------------------------------------
__global__ void fill_kernel(float* p, int n, float v) {
  int i = blockIdx.x * blockDim.x + threadIdx.x;
  if (i < n) p[i] = v;
}
__global__ void relu_kernel(float* p, int n) {
  int i = blockIdx.x * blockDim.x + threadIdx.x;
  if (i < n) p[i] = fmaxf(p[i], 0.f);
}
__global__ void deg_edge_kernel(const int* __restrict__ dst, float* deg, int E) {
  int e = blockIdx.x * blockDim.x + threadIdx.x;
  if (e < E) atomicAdd(&deg[dst[e]], 1.f);
}
__global__ void dinv_kernel(const float* deg, float* dinv, int n) {
  int i = blockIdx.x * blockDim.x + threadIdx.x;
  if (i < n) dinv[i] = rsqrtf(deg[i]);
}
__global__ void gcn_self_kernel(const float* __restrict__ xw,
                                const float* __restrict__ dinv,
                                const float* __restrict__ b,
                                float* __restrict__ out, int N, int F) {
  int idx = blockIdx.x * blockDim.x + threadIdx.x;
  if (idx >= N * F) return;
  int n = idx / F;
  int f = idx - n * F;
  out[idx] = xw[idx] * dinv[n] * dinv[n] + b[f];
}
__global__ void gcn_edge_kernel(const float* __restrict__ xw,
                                const int* __restrict__ src,
                                const int* __restrict__ dst,
                                const float* __restrict__ dinv,
                                float* __restrict__ out, int E, int F) {
  int idx = blockIdx.x * blockDim.x + threadIdx.x;
  if (idx >= E * F) return;
  int e = idx / F;
  int f = idx - e * F;
  int s = src[e], d = dst[e];
  atomicAdd(&out[(size_t)d * F + f], xw[(size_t)s * F + f] * dinv[s] * dinv[d]);
}
__global__ void smax_init_kernel(unsigned* u, int n) {
  int i = blockIdx.x * blockDim.x + threadIdx.x;
  if (i < n) u[i] = ord_enc(-3.402823466e38f);
}
__global__ void smax_scatter_kernel(const float* __restrict__ x,
                                    const int* __restrict__ seg,
                                    unsigned* __restrict__ out, int rows, int F) {
  int idx = blockIdx.x * blockDim.x + threadIdx.x;
  if (idx >= rows * F) return;
  int r = idx / F;
  int f = idx - r * F;
  atomicMax(&out[(size_t)seg[r] * F + f], ord_enc(x[idx]));
}
__global__ void smax_decode_kernel(unsigned* u, int n) {
  int i = blockIdx.x * blockDim.x + threadIdx.x;
  if (i < n) {
    float f = ord_dec(u[i]);
    ((float*)u)[i] = f;
  }
}
__global__ __launch_bounds__(32) void rownorm_kernel(const float* __restrict__ x,
                                                     float* __restrict__ out,
                                                     int C) {
  int row = blockIdx.x;
  int lane = threadIdx.x;
  const float* xr = x + (size_t)row * C;
  float ss = 0.f;
  for (int c = lane; c < C; c += 32) {
    float v = xr[c];
    ss += v * v;
  }
#pragma unroll
  for (int m = 1; m <= 16; m <<= 1) ss += __shfl_xor(ss, m, 32);
  float s = 1.f / fmaxf(sqrtf(ss), 1e-12f);
  for (int c = lane; c < C; c += 32) out[(size_t)row * C + c] = xr[c] * s;
}

// ---------------------------------------------------------------------------
// Host orchestration
// ---------------------------------------------------------------------------
extern "C" void kernel_launch(void* const* d_in, const int* in_sizes, int n_in,
                              void* d_out, int out_size, void* d_ws,
                              size_t ws_size, hipStream_t stream) {
  (void)in_sizes; (void)n_in; (void)out_size; (void)ws_size;
  const int L = 2048, NN = 2048, EE = 8192, NG = 32;

#define FP(i) ((const float*)d_in[(i)])
#define IP(i) ((const int*)d_in[(i)])

  // --- workspace bump allocator
  char* wp = (char*)d_ws;
  auto bump = [&](size_t nflt) -> float* {
    float* r = (float*)wp;
    wp += ((nflt * sizeof(float) + 255) & ~(size_t)255);
    return r;
  };
  float* d1 = bump((size_t)L * 128);
  float* d2 = bump((size_t)L * 128);
  float* oep = bump((size_t)L * 128);
  float* oes = bump((size_t)L * 128);
  float* trg = bump((size_t)L * 128);
  float* Qp = bump((size_t)L * 128);
  float* Kp = bump((size_t)L * 128);
  float* Vp = bump((size_t)L * 128);
  float* tO = bump((size_t)L * 128);
  float* tR = bump((size_t)L * 128);
  float* pfH = bump((size_t)L * 256);
  float* ci = bump((size_t)L * 128);
  float* colB = bump((size_t)L * 1152);
  float* wrB = bump((size_t)1152 * 256);
  float* convOut = bump((size_t)L * 256);
  float* deg = bump(NN);
  float* dinv = bump(NN);
  float* xw = bump((size_t)NN * 312);
  float* hA = bump((size_t)NN * 312);
  float* hB = bump((size_t)NN * 312);
  unsigned* poolU = (unsigned*)bump((size_t)NG * 312);
  float* t156 = bump((size_t)NG * 156);
  float* g1b = bump((size_t)NG * 128);
  float* g2b = bump((size_t)NG * 128);
  unsigned* iddU = (unsigned*)bump((size_t)NG * 128);
  unsigned* idd1U = (unsigned*)bump((size_t)NG * 128);
  float* cva = bump((size_t)NG * 512);
  float* cvb = bump((size_t)NG * 256);
  float* cvv = bump((size_t)NG * 128);
  float* dcv1 = bump((size_t)NG * 256);
  float* dcv2 = bump((size_t)NG * 512);
  float* hh1 = bump((size_t)NG * 512);
  float* hh2 = bump((size_t)NG * 128);

  auto gemm = [&](const float* A, const float* B, const float* bias, float* C,
                  int M, int N, int K, int act, const float* Cin) {
    int tiles = ((M + 15) / 16) * ((N + 15) / 16);
    int blocks = (tiles + GEMM_WAVES - 1) / GEMM_WAVES;
    gemm_wmma_kernel<<<blocks, 256, 0, stream>>>(A, B, bias, Cin, C, M, N, K,
                                                 act);
  };

  auto attention = [&](const float* q, const float* kv, int pb, float* dst) {
    gemm(q, FP(pb + 0), FP(pb + 1), Qp, L, 128, 128, 0, nullptr);
    gemm(kv, FP(pb + 2), FP(pb + 3), Kp, L, 128, 128, 0, nullptr);
    gemm(kv, FP(pb + 4), FP(pb + 5), Vp, L, 128, 128, 0, nullptr);
    attn_flash_kernel<<<dim3(L / 16, 8), 32, 0, stream>>>(Qp, Kp, Vp, tO, L, L);
    gemm(tO, FP(pb + 6), FP(pb + 7), dst, L, 128, 128, 0, nullptr);
  };

  auto encoder = [&](const float* din, int base, float* outbuf) {
    gemm(din, FP(base + 0), FP(base + 1), ci, L, 128, 128, 0, nullptr);
    for (int c = 0; c < 3; ++c) {
      im2col_kernel<<<(L * 1152 + 255) / 256, 256, 0, stream>>>(ci, colB, L);
      repack_w_kernel<<<(256 * 128 * 9 + 255) / 256, 256, 0, stream>>>(
          FP(base + 2 + c * 2), wrB);
      gemm(colB, wrB, FP(base + 3 + c * 2), convOut, L, 256, 1152, 0, nullptr);
      glu_res_kernel<<<(L * 128 + 255) / 256, 256, 0, stream>>>(convOut, ci, L);
    }
    layernorm_kernel<<<L, 32, 0, stream>>>(ci, nullptr, FP(base + 8),
                                           FP(base + 9), outbuf);
  };

  auto decoder = [&](const float* din, const float* src, int base) {
    gemm(din, FP(base + 0), FP(base + 1), trg, L, 128, 128, 0, nullptr);
    for (int l = 0; l < 3; ++l) {
      int lb = base + 2 + l * 22;
      const float* lg = FP(lb + 0);
      const float* lbias = FP(lb + 1);
      attention(trg, trg, lb + 2, tR);                       // self-attn
      layernorm_kernel<<<L, 32, 0, stream>>>(trg, tR, lg, lbias, trg);
      attention(trg, src, lb + 10, tR);                      // cross-attn
      layernorm_kernel<<<L, 32, 0, stream>>>(trg, tR, lg, lbias, trg);
      gemm(trg, FP(lb + 18), FP(lb + 19), pfH, L, 256, 128, 1, nullptr);
      gemm(pfH, FP(lb + 20), FP(lb + 21), tR, L, 128, 256, 0, nullptr);
      layernorm_kernel<<<L, 32, 0, stream>>>(trg, tR, lg, lbias, trg);
    }
  };

  auto segmax = [&](const float* x, const int* seg, int rows, int F,
                    unsigned* u) -> float* {
    smax_init_kernel<<<(NG * F + 255) / 256, 256, 0, stream>>>(u, NG * F);
    smax_scatter_kernel<<<(rows * F + 255) / 256, 256, 0, stream>>>(x, seg, u,
                                                                    rows, F);
    smax_decode_kernel<<<(NG * F + 255) / 256, 256, 0, stream>>>(u, NG * F);
    return (float*)u;
  };

  auto drug = [&](const float* x, const int* ei, const int* batch, float* gout) {
    const int* src = ei;
    const int* dst = ei + EE;
    fill_kernel<<<(NN + 255) / 256, 256, 0, stream>>>(deg, NN, 1.0f);
    deg_edge_kernel<<<(EE + 255) / 256, 256, 0, stream>>>(dst, deg, EE);
    dinv_kernel<<<(NN + 255) / 256, 256, 0, stream>>>(deg, dinv, NN);
    auto layer = [&](const float* hin, int Fin, int Fout, int wi, int bi,
                     float* hout) {
      gemm(hin, FP(wi), nullptr, xw, NN, Fout, Fin, 0, nullptr);
      gcn_self_kernel<<<(NN * Fout + 255) / 256, 256, 0, stream>>>(
          xw, dinv, FP(bi), hout, NN, Fout);
      gcn_edge_kernel<<<(EE * Fout + 255) / 256, 256, 0, stream>>>(
          xw, src, dst, dinv, hout, EE, Fout);
      relu_kernel<<<(NN * Fout + 255) / 256, 256, 0, stream>>>(hout, NN * Fout);
    };
    layer(x, 78, 78, 7, 8, hA);
    layer(hA, 78, 156, 9, 10, hB);
    layer(hB, 156, 312, 11, 12, hA);
    float* pool = segmax(hA, batch, NN, 312, poolU);
    gemm(pool, FP(13), FP(14), t156, NG, 156, 312, 1, nullptr);
    gemm(t156, FP(15), FP(16), gout, NG, 128, 156, 0, nullptr);
  };

  // ---- top-level forward ----
  // shared token projection fc: [L,100] @ (100,128)
  gemm(FP(1), FP(5), FP(6), d1, L, 128, 100, 0, nullptr);
  gemm(FP(3), FP(5), FP(6), d2, L, 128, 100, 0, nullptr);

  encoder(d1, 39, oep);              // enc_prot(d1)
  decoder(d2, oep, 59);              // ods = dec_smi(d2, oep) -> trg
  float* idd1 = segmax(trg, IP(200), L, 128, idd1U);  // new_batch2
  encoder(d2, 49, oes);              // enc_smi(d2)
  decoder(d1, oes, 127);             // odp = dec_prot(d1, oes) -> trg
  float* idd = segmax(trg, IP(199), L, 128, iddU);    // new_batch1

  drug(FP(0), IP(195), IP(197), g1b);
  drug(FP(2), IP(196), IP(198), g2b);

  float* out = (float*)d_out;
  float* o_h1 = out;            // head(xc_f1): 32x2
  float* o_cv0 = out + 64;      // cv0: 32x954
  float* o_dcv = out + 30592;   // dcv: 32x954
  float* o_x1 = out + 61120;    // xc_f1: 32x128
  float* o_x2 = out + 65216;    // xc_f2: 32x128
  float* o_h2 = out + 69312;    // head(xc_f2): 32x2

  rownorm_kernel<<<NG, 32, 0, stream>>>(FP(4), o_cv0, 954);
  gemm(o_cv0, FP(17), FP(18), cva, NG, 512, 954, 1, nullptr);
  gemm(cva, FP(19), FP(20), cvb, NG, 256, 512, 1, nullptr);
  gemm(cvb, FP(21), FP(22), cvv, NG, 128, 256, 0, nullptr);
  gemm(cvv, FP(23), FP(24), dcv1, NG, 256, 128, 1, nullptr);
  gemm(dcv1, FP(25), FP(26), dcv2, NG, 512, 256, 1, nullptr);
  gemm(dcv2, FP(27), FP(28), o_dcv, NG, 954, 512, 0, nullptr);

  // xc_f1 = m1(g1)+m1(g2)+m1(idd)+m1(cv)  (bias added per call, matching ref)
  gemm(g1b, FP(29), FP(30), o_x1, NG, 128, 128, 0, nullptr);
  gemm(g2b, FP(29), FP(30), o_x1, NG, 128, 128, 0, o_x1);
  gemm(idd, FP(29), FP(30), o_x1, NG, 128, 128, 0, o_x1);
  gemm(cvv, FP(29), FP(30), o_x1, NG, 128, 128, 0, o_x1);
  // xc_f2 = m2(g1)+m2(g2)+m2(idd1)+m2(cv)
  gemm(g1b, FP(31), FP(32), o_x2, NG, 128, 128, 0, nullptr);
  gemm(g2b, FP(31), FP(32), o_x2, NG, 128, 128, 0, o_x2);
  gemm(idd1, FP(31), FP(32), o_x2, NG, 128, 128, 0, o_x2);
  gemm(cvv, FP(31), FP(32), o_x2, NG, 128, 128, 0, o_x2);

  auto head = [&](const float* xc, float* dst) {
    gemm(xc, FP(33), FP(34), hh1, NG, 512, 128, 1, nullptr);
    gemm(hh1, FP(35), FP(36), hh2, NG, 128, 512, 1, nullptr);
    gemm(hh2, FP(37), FP(38), dst, NG, 2, 128, 0, nullptr);
  };
  head(o_x1, o_h1);
  head(o_x2, o_h2);

#undef FP
#undef IP
}